// APPNP_28381143892760
// MI455X (gfx1250) — compile-verified
//
#include <hip/hip_runtime.h>

// APPNP on MI455X (gfx1250, wave32).
// Strategy: build CSR once (amortized over K=10 propagation steps), pull-based
// atomic-free aggregation (one wave per dst node, float2 per lane = one
// coalesced 256B row load per edge), and the (1-a)*agg + a*x combine done with
// V_WMMA_F32_16X16X4_F32 on 16x16 tiles (A = 0.9*selector, C = 0.1*x), which
// is exact in f32. Working set ~62MB fits in the 192MB L2, so all 10
// iterations run out of L2 (~3.5GB of L2 traffic total).

#define D_FEAT 64
#define K_STEPS 10

typedef __attribute__((ext_vector_type(2))) float v2f;
typedef __attribute__((ext_vector_type(8))) float v8f;

__global__ void k_zero_i32(int* __restrict__ p, int n) {
    int i = blockIdx.x * blockDim.x + threadIdx.x;
    if (i < n) p[i] = 0;
}

// deg[i] = number of edges whose destination (col) is i
__global__ void k_degree(const int* __restrict__ col, int* __restrict__ deg, int e) {
    int i = blockIdx.x * blockDim.x + threadIdx.x;
    if (i < e) atomicAdd(&deg[col[i]], 1);
}

__global__ void k_dinv(const int* __restrict__ deg, float* __restrict__ dinv, int n) {
    int i = blockIdx.x * blockDim.x + threadIdx.x;
    if (i < n) {
        int d = deg[i];
        dinv[i] = (d > 0) ? rsqrtf((float)d) : 0.0f;
    }
}

// Single-block exclusive scan of deg -> ptr (N+1), cursor copy for CSR build.
// N=100k, 1024 threads, ~98 elements each: runs once, off critical path.
__global__ void __launch_bounds__(1024) k_scan(const int* __restrict__ deg,
                                               int* __restrict__ ptr,
                                               int* __restrict__ cur, int n) {
    __shared__ int sh[1024];
    int t = threadIdx.x;
    int per = (n + 1023) >> 10;
    int b = t * per; if (b > n) b = n;
    int e = b + per; if (e > n) e = n;
    int s = 0;
    for (int i = b; i < e; ++i) s += deg[i];
    sh[t] = s;
    __syncthreads();
    int val = s;
    for (int off = 1; off < 1024; off <<= 1) {
        int tmp = (t >= off) ? sh[t - off] : 0;
        __syncthreads();
        sh[t] += tmp;
        __syncthreads();
    }
    int run = sh[t] - val;  // exclusive prefix of this thread's chunk
    for (int i = b; i < e; ++i) {
        ptr[i] = run;
        cur[i] = run;
        run += deg[i];
    }
    if (t == 1023) ptr[n] = sh[1023];
}

// Scatter edges into CSR slots; fold gcn_norm into per-edge weight.
__global__ void k_build(const int* __restrict__ row, const int* __restrict__ col,
                        const float* __restrict__ dinv, int* __restrict__ cur,
                        int* __restrict__ csrc, float* __restrict__ cnrm, int e) {
    int i = blockIdx.x * blockDim.x + threadIdx.x;
    if (i < e) {
        int s = row[i], d = col[i];
        int pos = atomicAdd(&cur[d], 1);
        csrc[pos] = s;
        cnrm[pos] = dinv[s] * dinv[d];
    }
}

// Pull aggregation: one wave32 per destination node; lane holds float2 so one
// edge = one global_load_b64 per lane = one coalesced 256B h-row per wave.
__global__ void __launch_bounds__(256) k_aggregate(
        const int* __restrict__ ptr, const int* __restrict__ csrc,
        const float* __restrict__ cnrm, const float* __restrict__ hsrc,
        float* __restrict__ agg, int n) {
    int wid = blockIdx.x * (blockDim.x >> 5) + (threadIdx.x >> 5);
    if (wid >= n) return;
    int lane = threadIdx.x & 31;
    int beg = ptr[wid];
    int end = ptr[wid + 1];
    const v2f* __restrict__ h2 = (const v2f*)hsrc;
    float ax = 0.0f, ay = 0.0f;
    for (int e = beg; e < end; ++e) {
        int s = csrc[e];
        float w = cnrm[e];
        v2f v = h2[(size_t)s * 32 + lane];
        ax = fmaf(w, v.x, ax);
        ay = fmaf(w, v.y, ay);
    }
    v2f acc; acc.x = ax; acc.y = ay;
    ((v2f*)agg)[(size_t)wid * 32 + lane] = acc;
}

// h = 0.9*agg + 0.1*x per 16(nodes)x16(features) tile via chained
// V_WMMA_F32_16X16X4_F32:  D = sum_j A_j(16x4) * B_j(4x16) + C
//   A_j[m,k] = 0.9 * delta(m, 4j+k)   (built from lane id, no memory)
//   B_j      = agg rows 4j..4j+3 of the tile
//   C        = 0.1 * x tile
// VGPR layouts per ISA 7.12.2 (wave32): C/D lanes0-15 -> M=vgpr, lanes16-31 ->
// M=vgpr+8; A/B K={0,1} in lanes0-15, K={2,3} in lanes16-31 across 2 VGPRs.
__global__ void __launch_bounds__(256) k_combine_wmma(
        const float* __restrict__ agg, const float* __restrict__ x,
        float* __restrict__ out, int ntiles) {
    int tid = blockIdx.x * (blockDim.x >> 5) + (threadIdx.x >> 5);
    if (tid >= ntiles) return;
    int lane = threadIdx.x & 31;
    int half = lane >> 4;
    int lh   = lane & 15;
    int ntile = tid >> 2;   // 16-node group
    int ftile = tid & 3;    // 16-feature group (D=64 -> 4 tiles)
    size_t base = (size_t)ntile * 16 * D_FEAT + (size_t)ftile * 16;

    v8f c;
#pragma unroll
    for (int v = 0; v < 8; ++v) {
        int m = v + 8 * half;
        c[v] = 0.1f * x[base + (size_t)m * D_FEAT + lh];
    }
    int k0 = half ? 2 : 0;
#pragma unroll
    for (int j = 0; j < 4; ++j) {
        v2f b;
        b.x = agg[base + (size_t)(4 * j + k0    ) * D_FEAT + lh];
        b.y = agg[base + (size_t)(4 * j + k0 + 1) * D_FEAT + lh];
        v2f a;
        a.x = (lh == 4 * j + k0    ) ? 0.9f : 0.0f;
        a.y = (lh == 4 * j + k0 + 1) ? 0.9f : 0.0f;
        // (neg_a, A, neg_b, B, c_mod, C, reuse_a, reuse_b)
        c = __builtin_amdgcn_wmma_f32_16x16x4_f32(false, a, false, b,
                                                  (short)0, c, false, false);
    }
#pragma unroll
    for (int v = 0; v < 8; ++v) {
        int m = v + 8 * half;
        out[base + (size_t)m * D_FEAT + lh] = c[v];
    }
}

// Scalar tail for N % 16 nodes (N=100000 -> unused, kept for robustness).
__global__ void k_combine_tail(const float* __restrict__ agg,
                               const float* __restrict__ x,
                               float* __restrict__ out, int n0, int n) {
    int i = n0 * D_FEAT + blockIdx.x * blockDim.x + threadIdx.x;
    if (i < n * D_FEAT) out[i] = 0.9f * agg[i] + 0.1f * x[i];
}

extern "C" void kernel_launch(void* const* d_in, const int* in_sizes, int n_in,
                              void* d_out, int out_size, void* d_ws, size_t ws_size,
                              hipStream_t stream) {
    const float* x = (const float*)d_in[0];
    const int* ei  = (const int*)d_in[1];
    const int E = in_sizes[1] / 2;
    const int N = in_sizes[0] / D_FEAT;
    const int* row = ei;       // edge_index[0] = sources
    const int* col = ei + E;   // edge_index[1] = destinations

    // Bump allocator over preallocated workspace (256B aligned slices).
    char* ws = (char*)d_ws;
    size_t off = 0;
    auto alloc = [&](size_t bytes) -> char* {
        char* p = ws + off;
        off = (off + bytes + 255) & ~(size_t)255;
        return p;
    };
    int*   deg  = (int*)  alloc((size_t)N * 4);
    float* dinv = (float*)alloc((size_t)N * 4);
    int*   ptr  = (int*)  alloc((size_t)(N + 1) * 4);
    int*   cur  = (int*)  alloc((size_t)N * 4);
    int*   csrc = (int*)  alloc((size_t)E * 4);
    float* cnrm = (float*)alloc((size_t)E * 4);
    float* agg  = (float*)alloc((size_t)N * D_FEAT * 4);
    float* h    = (float*)alloc((size_t)N * D_FEAT * 4);

    // ---- one-time graph preprocessing (amortized over K=10 steps) ----
    k_zero_i32<<<(N + 255) / 256, 256, 0, stream>>>(deg, N);
    k_degree  <<<(E + 255) / 256, 256, 0, stream>>>(col, deg, E);
    k_dinv    <<<(N + 255) / 256, 256, 0, stream>>>(deg, dinv, N);
    k_scan    <<<1, 1024, 0, stream>>>(deg, ptr, cur, N);
    k_build   <<<(E + 255) / 256, 256, 0, stream>>>(row, col, dinv, cur, csrc, cnrm, E);

    // ---- K propagation steps, all traffic served out of L2 ----
    const int aggBlocks = (N + 7) / 8;                 // 8 waves / block
    const int nfull     = (N / 16) * 16;
    const int ntiles    = (N / 16) * (D_FEAT / 16);
    const int combBlocks = (ntiles + 7) / 8;
    const int rem = N - nfull;

    for (int t = 0; t < K_STEPS; ++t) {
        const float* hs = (t == 0) ? x : h;
        float* ho = (t == K_STEPS - 1) ? (float*)d_out : h;
        k_aggregate<<<aggBlocks, 256, 0, stream>>>(ptr, csrc, cnrm, hs, agg, N);
        if (ntiles > 0)
            k_combine_wmma<<<combBlocks, 256, 0, stream>>>(agg, x, ho, ntiles);
        if (rem > 0)
            k_combine_tail<<<(rem * D_FEAT + 255) / 256, 256, 0, stream>>>(
                agg, x, ho, nfull, N);
    }
}